// EGNNLayer_41669772706275
// MI455X (gfx1250) — compile-verified
//
#include <hip/hip_runtime.h>
#include <hip/hip_bf16.h>

// ---------------------------------------------------------------------------
// EGNN layer, fp32 WMMA implementation for gfx1250 (MI455X).
//   kernel 1: zero agg workspace
//   kernel 2: per 16-edge tile: async-gather h[s],h[r] to LDS, dist ->
//             MLP(W1,silu,W2,silu) via V_WMMA_F32_16X16X4_F32 ->
//             native f32 atomic scatter-add into agg (L2-resident)
//   kernel 3: per 16-node tile: [h|agg] -> MLP(U1,silu,U2) + h -> out
// ---------------------------------------------------------------------------

typedef float v2f __attribute__((ext_vector_type(2)));
typedef float v8f __attribute__((ext_vector_type(8)));

#define EGNN_N 50000
#define EGNN_E 800000
#define EGNN_D 128
#define WAVES  4          // waves per block
#define K1PAD  260        // 2D+1 = 257 padded to multiple of 4
#define MSTR   130        // Mtile row stride (bank-conflict spread)

__device__ __forceinline__ float silu_f(float x) {
    // x * sigmoid(x); v_rcp_f32 instead of IEEE divide (saves ~10 VALU/elt)
    return x * __builtin_amdgcn_rcpf(1.0f + __expf(-x));
}

__device__ __forceinline__ unsigned lds_off(const void* p) {
    // generic LDS pointer: addr[31:0] is the wave-relative LDS byte offset
    return (unsigned)(unsigned long long)p;
}

__device__ __forceinline__ void async_gather_b128(unsigned lds_dst,
                                                  unsigned byte_off,
                                                  const float* base) {
    // GLOBAL_LOAD_ASYNC_TO_LDS_B128, GVS mode: SGPR base + per-lane i32 offset
    asm volatile("global_load_async_to_lds_b128 %0, %1, %2"
                 :: "v"(lds_dst), "v"(byte_off), "s"(base) : "memory");
}

__device__ __forceinline__ void wait_async0() {
    asm volatile("s_wait_asynccnt 0x0" ::: "memory");
}

__global__ void egnn_zero_kernel(float* __restrict__ p, int n) {
    int i = blockIdx.x * blockDim.x + threadIdx.x;
    if (i < n) p[i] = 0.0f;
}

// ------------------------------ edge kernel --------------------------------
__global__ __launch_bounds__(WAVES * 32)
void egnn_edge_kernel(const float* __restrict__ h,
                      const float* __restrict__ coords,
                      const long long* __restrict__ ei,   // [2][E] int64
                      const float* __restrict__ W1,       // [128][257]
                      const float* __restrict__ b1,
                      const float* __restrict__ W2,       // [128][128]
                      const float* __restrict__ b2,
                      float* __restrict__ agg)            // [N][128]
{
    // fragment-swizzled B panels: Bs[kk*2+half][n] = (Wt[k0+2h][n], Wt[k0+2h+1][n])
    __shared__ v2f Bs1[65 * 2 * 128];                 // 133120 B
    __shared__ v2f Bs2[32 * 2 * 128];                 //  65536 B
    __shared__ __align__(16) float Atile[WAVES][16 * K1PAD];  // 66560 B
    __shared__ __align__(16) float Mtile[WAVES][16 * MSTR];   // 33280 B
    __shared__ float b1s[128], b2s[128];
    __shared__ int   Sidx[WAVES][16], Ridx[WAVES][16];

    const int tid = threadIdx.x;

    // stage W1^T (K=257 padded to 260, zero-filled) in fragment order
    for (int i = tid; i < 65 * 2 * 128; i += blockDim.x) {
        int kk = i >> 8, rem = i & 255;
        int hfi = rem >> 7, n = rem & 127;
        int k0 = kk * 4 + hfi * 2;
        float x = (k0     < 257) ? W1[n * 257 + k0]     : 0.0f;
        float y = (k0 + 1 < 257) ? W1[n * 257 + k0 + 1] : 0.0f;
        v2f b; b.x = x; b.y = y;
        Bs1[i] = b;
    }
    // stage W2^T
    for (int i = tid; i < 32 * 2 * 128; i += blockDim.x) {
        int kk = i >> 8, rem = i & 255;
        int hfi = rem >> 7, n = rem & 127;
        int k0 = kk * 4 + hfi * 2;
        v2f b; b.x = W2[n * 128 + k0]; b.y = W2[n * 128 + k0 + 1];
        Bs2[i] = b;
    }
    if (tid < 128) { b1s[tid] = b1[tid]; b2s[tid] = b2[tid]; }
    __syncthreads();

    const int wave = tid >> 5, lane = tid & 31;
    const int hf = lane >> 4, ln = lane & 15;
    float* At = Atile[wave];
    float* Mt = Mtile[wave];
    int*   Si = Sidx[wave];
    int*   Ri = Ridx[wave];
    const unsigned AtL = lds_off(At);

    const int wgid    = blockIdx.x * WAVES + wave;
    const int wstride = gridDim.x * WAVES;
    const int ntiles  = EGNN_E / 16;   // 50000

    for (int t = wgid; t < ntiles; t += wstride) {
        const int e0 = t * 16;

        // edge endpoint indices (wave-synchronous LDS exchange)
        if (hf == 0) Si[ln] = (int)ei[e0 + ln];
        else         Ri[ln] = (int)ei[(long long)EGNN_E + e0 + ln];

        // distance column (k = 256) + zero pad (k = 257..259)
        if (hf == 0) {
            int s = Si[ln], r = Ri[ln];
            float dx = coords[s * 3 + 0] - coords[r * 3 + 0];
            float dy = coords[s * 3 + 1] - coords[r * 3 + 1];
            float dz = coords[s * 3 + 2] - coords[r * 3 + 2];
            At[ln * K1PAD + 256] = __fsqrt_rn(dx * dx + dy * dy + dz * dz);
            At[ln * K1PAD + 257] = 0.0f;
            At[ln * K1PAD + 258] = 0.0f;
            At[ln * K1PAD + 259] = 0.0f;
        }

        // async-gather h[sender] | h[receiver] straight into LDS (no VGPR staging)
        #pragma unroll 4
        for (int j = 0; j < 32; ++j) {
            int idx  = j * 32 + lane;          // 0..1023 float4 slots
            int row  = idx >> 6;
            int c4   = idx & 63;
            int node = (c4 < 32) ? Si[row] : Ri[row];
            int col  = (c4 & 31) * 4;
            unsigned dst  = AtL + (unsigned)(row * K1PAD + ((c4 < 32) ? col : 128 + col)) * 4u;
            unsigned goff = (unsigned)(node * 512 + col * 4);
            async_gather_b128(dst, goff, h);
        }
        wait_async0();

        // GEMM1: [16,260] x [260,128]; 4 N-tiles per K-sweep (A-frag reuse, 4x ILP)
        for (int np = 0; np < 2; ++np) {
            v8f acc0 = {}, acc1 = {}, acc2 = {}, acc3 = {};
            const int nb = np * 64 + ln;
            #pragma unroll 5
            for (int kk = 0; kk < 65; ++kk) {
                v2f a = *(const v2f*)(At + ln * K1PAD + kk * 4 + hf * 2);
                const v2f* brow = &Bs1[(kk * 2 + hf) * 128 + nb];
                v2f b0 = brow[0], b1v = brow[16], b2v = brow[32], b3v = brow[48];
                acc0 = __builtin_amdgcn_wmma_f32_16x16x4_f32(false, a, false, b0,  (short)0, acc0, false, false);
                acc1 = __builtin_amdgcn_wmma_f32_16x16x4_f32(false, a, false, b1v, (short)0, acc1, false, false);
                acc2 = __builtin_amdgcn_wmma_f32_16x16x4_f32(false, a, false, b2v, (short)0, acc2, false, false);
                acc3 = __builtin_amdgcn_wmma_f32_16x16x4_f32(false, a, false, b3v, (short)0, acc3, false, false);
            }
            v8f* accs[4] = { &acc0, &acc1, &acc2, &acc3 };
            #pragma unroll
            for (int u = 0; u < 4; ++u) {
                const int n = nb + u * 16;
                float bias = b1s[n];
                v8f a8 = *accs[u];
                #pragma unroll
                for (int r = 0; r < 8; ++r) {
                    float x = a8[r] + bias;
                    Mt[(r + 8 * hf) * MSTR + n] = silu_f(x);
                }
            }
        }

        // GEMM2: [16,128] x [128,128], + b2, SiLU, atomic scatter to receiver
        for (int np = 0; np < 2; ++np) {
            v8f acc0 = {}, acc1 = {}, acc2 = {}, acc3 = {};
            const int nb = np * 64 + ln;
            #pragma unroll 4
            for (int kk = 0; kk < 32; ++kk) {
                v2f a = *(const v2f*)(Mt + ln * MSTR + kk * 4 + hf * 2);
                const v2f* brow = &Bs2[(kk * 2 + hf) * 128 + nb];
                v2f b0 = brow[0], b1v = brow[16], b2v = brow[32], b3v = brow[48];
                acc0 = __builtin_amdgcn_wmma_f32_16x16x4_f32(false, a, false, b0,  (short)0, acc0, false, false);
                acc1 = __builtin_amdgcn_wmma_f32_16x16x4_f32(false, a, false, b1v, (short)0, acc1, false, false);
                acc2 = __builtin_amdgcn_wmma_f32_16x16x4_f32(false, a, false, b2v, (short)0, acc2, false, false);
                acc3 = __builtin_amdgcn_wmma_f32_16x16x4_f32(false, a, false, b3v, (short)0, acc3, false, false);
            }
            v8f* accs[4] = { &acc0, &acc1, &acc2, &acc3 };
            #pragma unroll
            for (int u = 0; u < 4; ++u) {
                const int n = nb + u * 16;
                float bias = b2s[n];
                v8f a8 = *accs[u];
                #pragma unroll
                for (int r = 0; r < 8; ++r) {
                    float x = silu_f(a8[r] + bias);
                    int node = Ri[r + 8 * hf];
                    unsafeAtomicAdd(agg + node * 128 + n, x);
                }
            }
        }
    }
}

// ------------------------------ node kernel --------------------------------
__global__ __launch_bounds__(WAVES * 32)
void egnn_node_kernel(const float* __restrict__ h,
                      const float* __restrict__ agg,
                      const float* __restrict__ U1,   // [128][256]
                      const float* __restrict__ c1,
                      const float* __restrict__ U2,   // [128][128]
                      const float* __restrict__ c2,
                      float* __restrict__ out)
{
    __shared__ v2f Bs1[64 * 2 * 128];                 // 131072 B
    __shared__ v2f Bs2[32 * 2 * 128];                 //  65536 B
    __shared__ __align__(16) float Atile[WAVES][16 * K1PAD];
    __shared__ __align__(16) float Mtile[WAVES][16 * MSTR];
    __shared__ float c1s[128], c2s[128];

    const int tid = threadIdx.x;
    for (int i = tid; i < 64 * 2 * 128; i += blockDim.x) {
        int kk = i >> 8, rem = i & 255;
        int hfi = rem >> 7, n = rem & 127;
        int k0 = kk * 4 + hfi * 2;
        v2f b; b.x = U1[n * 256 + k0]; b.y = U1[n * 256 + k0 + 1];
        Bs1[i] = b;
    }
    for (int i = tid; i < 32 * 2 * 128; i += blockDim.x) {
        int kk = i >> 8, rem = i & 255;
        int hfi = rem >> 7, n = rem & 127;
        int k0 = kk * 4 + hfi * 2;
        v2f b; b.x = U2[n * 128 + k0]; b.y = U2[n * 128 + k0 + 1];
        Bs2[i] = b;
    }
    if (tid < 128) { c1s[tid] = c1[tid]; c2s[tid] = c2[tid]; }
    __syncthreads();

    const int wave = tid >> 5, lane = tid & 31;
    const int hf = lane >> 4, ln = lane & 15;
    float* At = Atile[wave];
    float* Mt = Mtile[wave];
    const unsigned AtL = lds_off(At);

    const int wgid    = blockIdx.x * WAVES + wave;
    const int wstride = gridDim.x * WAVES;
    const int ntiles  = EGNN_N / 16;   // 3125

    for (int t = wgid; t < ntiles; t += wstride) {
        const int n0 = t * 16;

        // async-gather [h | agg] into 16x256 tile (two uniform bases)
        #pragma unroll 4
        for (int j = 0; j < 16; ++j) {
            int idx = j * 32 + lane;           // 0..511 slots of h half
            int row = idx >> 5, c4 = idx & 31;
            unsigned dst  = AtL + (unsigned)(row * K1PAD + c4 * 4) * 4u;
            unsigned goff = (unsigned)((n0 + row) * 512 + c4 * 16);
            async_gather_b128(dst, goff, h);
        }
        #pragma unroll 4
        for (int j = 0; j < 16; ++j) {
            int idx = j * 32 + lane;           // agg half
            int row = idx >> 5, c4 = idx & 31;
            unsigned dst  = AtL + (unsigned)(row * K1PAD + 128 + c4 * 4) * 4u;
            unsigned goff = (unsigned)((n0 + row) * 512 + c4 * 16);
            async_gather_b128(dst, goff, agg);
        }
        wait_async0();

        // GEMM1: [16,256] x [256,128]; 4 N-tiles per K-sweep
        for (int np = 0; np < 2; ++np) {
            v8f acc0 = {}, acc1 = {}, acc2 = {}, acc3 = {};
            const int nb = np * 64 + ln;
            #pragma unroll 4
            for (int kk = 0; kk < 64; ++kk) {
                v2f a = *(const v2f*)(At + ln * K1PAD + kk * 4 + hf * 2);
                const v2f* brow = &Bs1[(kk * 2 + hf) * 128 + nb];
                v2f b0 = brow[0], b1v = brow[16], b2v = brow[32], b3v = brow[48];
                acc0 = __builtin_amdgcn_wmma_f32_16x16x4_f32(false, a, false, b0,  (short)0, acc0, false, false);
                acc1 = __builtin_amdgcn_wmma_f32_16x16x4_f32(false, a, false, b1v, (short)0, acc1, false, false);
                acc2 = __builtin_amdgcn_wmma_f32_16x16x4_f32(false, a, false, b2v, (short)0, acc2, false, false);
                acc3 = __builtin_amdgcn_wmma_f32_16x16x4_f32(false, a, false, b3v, (short)0, acc3, false, false);
            }
            v8f* accs[4] = { &acc0, &acc1, &acc2, &acc3 };
            #pragma unroll
            for (int u = 0; u < 4; ++u) {
                const int n = nb + u * 16;
                float bias = c1s[n];
                v8f a8 = *accs[u];
                #pragma unroll
                for (int r = 0; r < 8; ++r) {
                    float x = a8[r] + bias;
                    Mt[(r + 8 * hf) * MSTR + n] = silu_f(x);
                }
            }
        }

        // GEMM2: [16,128] x [128,128], + c2, + residual h -> out
        for (int np = 0; np < 2; ++np) {
            v8f acc0 = {}, acc1 = {}, acc2 = {}, acc3 = {};
            const int nb = np * 64 + ln;
            #pragma unroll 4
            for (int kk = 0; kk < 32; ++kk) {
                v2f a = *(const v2f*)(Mt + ln * MSTR + kk * 4 + hf * 2);
                const v2f* brow = &Bs2[(kk * 2 + hf) * 128 + nb];
                v2f b0 = brow[0], b1v = brow[16], b2v = brow[32], b3v = brow[48];
                acc0 = __builtin_amdgcn_wmma_f32_16x16x4_f32(false, a, false, b0,  (short)0, acc0, false, false);
                acc1 = __builtin_amdgcn_wmma_f32_16x16x4_f32(false, a, false, b1v, (short)0, acc1, false, false);
                acc2 = __builtin_amdgcn_wmma_f32_16x16x4_f32(false, a, false, b2v, (short)0, acc2, false, false);
                acc3 = __builtin_amdgcn_wmma_f32_16x16x4_f32(false, a, false, b3v, (short)0, acc3, false, false);
            }
            v8f* accs[4] = { &acc0, &acc1, &acc2, &acc3 };
            #pragma unroll
            for (int u = 0; u < 4; ++u) {
                const int n = nb + u * 16;
                float bias = c2s[n];
                v8f a8 = *accs[u];
                #pragma unroll
                for (int r = 0; r < 8; ++r) {
                    int node = n0 + r + 8 * hf;
                    out[node * 128 + n] = h[node * 128 + n] + a8[r] + bias;
                }
            }
        }
    }
}

// ------------------------------ launcher -----------------------------------
extern "C" void kernel_launch(void* const* d_in, const int* in_sizes, int n_in,
                              void* d_out, int out_size, void* d_ws, size_t ws_size,
                              hipStream_t stream) {
    const float*     h      = (const float*)d_in[0];
    const float*     coords = (const float*)d_in[1];
    const long long* ei     = (const long long*)d_in[2];
    const float*     W1     = (const float*)d_in[3];
    const float*     b1     = (const float*)d_in[4];
    const float*     W2     = (const float*)d_in[5];
    const float*     b2     = (const float*)d_in[6];
    const float*     U1     = (const float*)d_in[7];
    const float*     c1     = (const float*)d_in[8];
    const float*     U2     = (const float*)d_in[9];
    const float*     c2     = (const float*)d_in[10];
    float* out = (float*)d_out;
    float* agg = (float*)d_ws;   // N*D fp32 scratch, L2-resident

    const int nd = EGNN_N * EGNN_D;
    egnn_zero_kernel<<<(nd + 255) / 256, 256, 0, stream>>>(agg, nd);
    egnn_edge_kernel<<<1024, WAVES * 32, 0, stream>>>(h, coords, ei,
                                                      W1, b1, W2, b2, agg);
    egnn_node_kernel<<<512, WAVES * 32, 0, stream>>>(h, agg, U1, c1, U2, c2, out);
}